// ImprovedCustomMaskRCNN_26379689132283
// MI455X (gfx1250) — compile-verified
//
#include <hip/hip_runtime.h>
#include <hip/hip_bf16.h>

#define MASK_S   28
#define SS       (MASK_S * MASK_S)   // 784
#define IMG_H    520
#define IMG_W    704
#define NUM_GT   128
#define NUM_PROP 512

typedef float v2f __attribute__((ext_vector_type(2)));
typedef float v4f __attribute__((ext_vector_type(4)));
typedef float v8f __attribute__((ext_vector_type(8)));

// ---------------------------------------------------------------------------
// Kernel 1: box IoU (512 proposals x 128 GT), argmax match, pos flag.
// One thread per proposal; GT boxes staged in LDS (2 KB).
// ---------------------------------------------------------------------------
__global__ void __launch_bounds__(256)
iou_match_kernel(const float* __restrict__ proposals,
                 const float* __restrict__ gt_boxes,
                 int* __restrict__ matched,
                 float* __restrict__ pos)
{
    __shared__ float sgt[NUM_GT * 4];
    const int tid = threadIdx.x;
    for (int i = tid; i < NUM_GT * 4; i += blockDim.x) sgt[i] = gt_boxes[i];
    __syncthreads();

    const int p = blockIdx.x * blockDim.x + tid;
    if (p >= NUM_PROP) return;

    const float ax1 = proposals[p * 4 + 0];
    const float ay1 = proposals[p * 4 + 1];
    const float ax2 = proposals[p * 4 + 2];
    const float ay2 = proposals[p * 4 + 3];
    const float area_a = (ax2 - ax1) * (ay2 - ay1);

    float best = -1.0f;
    int   bidx = 0;
    for (int g = 0; g < NUM_GT; ++g) {
        const float bx1 = sgt[g * 4 + 0];
        const float by1 = sgt[g * 4 + 1];
        const float bx2 = sgt[g * 4 + 2];
        const float by2 = sgt[g * 4 + 3];
        const float area_b = (bx2 - bx1) * (by2 - by1);
        const float lx = fmaxf(ax1, bx1), ly = fmaxf(ay1, by1);
        const float rx = fminf(ax2, bx2), ry = fminf(ay2, by2);
        const float w = fmaxf(rx - lx, 0.0f), h = fmaxf(ry - ly, 0.0f);
        const float inter = w * h;
        const float iou = inter / (area_a + area_b - inter);
        if (iou > best) { best = iou; bidx = g; }   // first-max, matches jnp.argmax
    }
    matched[p] = bidx;
    pos[p]     = (best > 0.5f) ? 1.0f : 0.0f;
}

// ---------------------------------------------------------------------------
// Kernel 2: one wave32 block per proposal. Bilinear crop-resize gathers +
// BCE, reduced in full fp32 with V_WMMA_F32_16X16X4_F32 (A carries 64
// values/iter, B = all-ones 4x16). Branch-free tail masking keeps EXEC
// all-ones at every point; two independent accumulators per reduction
// break the WMMA RAW chain for ILP.
// ---------------------------------------------------------------------------
__device__ __forceinline__ float bce_pixel(int idx,
                                           const float* __restrict__ m,
                                           const float* __restrict__ lg1,
                                           int x1, int y1, int cwi, int chi,
                                           float cw, float ch, float posw)
{
    const int i = idx / MASK_S, j = idx % MASK_S;
    const float sy = fminf(fmaxf((i + 0.5f) * ch * (1.0f / MASK_S) - 0.5f, 0.0f), ch - 1.0f);
    const float sx = fminf(fmaxf((j + 0.5f) * cw * (1.0f / MASK_S) - 0.5f, 0.0f), cw - 1.0f);
    const float y0f = floorf(sy), x0f = floorf(sx);
    const float wy = sy - y0f,   wx = sx - x0f;
    const int y0i = (int)y0f,    x0i = (int)x0f;
    const int y1i = min(y0i + 1, chi - 1);
    const int x1i = min(x0i + 1, cwi - 1);
    const float* row0 = m + (size_t)(y0i + y1) * IMG_W;
    const float* row1 = m + (size_t)(y1i + y1) * IMG_W;
    const float v00 = row0[x0i + x1], v01 = row0[x1i + x1];
    const float v10 = row1[x0i + x1], v11 = row1[x1i + x1];
    const float top = (1.0f - wx) * v00 + wx * v01;
    const float bot = (1.0f - wx) * v10 + wx * v11;
    const float t   = (1.0f - wy) * top + wy * bot;
    const float l   = lg1[idx];
    const float bce = fmaxf(l, 0.0f) - l * t + log1pf(expf(-fabsf(l)));
    return bce * posw;
}

__global__ void __launch_bounds__(32)
mask_loss_kernel(const float* __restrict__ gt_boxes,
                 const float* __restrict__ gt_masks,
                 const float* __restrict__ mask_logits,
                 const int*   __restrict__ matched,
                 const float* __restrict__ pos,
                 float* __restrict__ bsum,   // per-proposal sum(bce * pos)
                 float* __restrict__ qsum)   // per-proposal sum(logits^2)
{
    const int p    = blockIdx.x;
    const int lane = threadIdx.x;

    const int   gidx = matched[p];
    const float posw = pos[p];

    // Matched GT box -> integer crop window (reference truncation semantics;
    // GT coords are positive so (int) == astype(int32)).
    const float b0 = gt_boxes[gidx * 4 + 0];
    const float b1 = gt_boxes[gidx * 4 + 1];
    const float b2 = gt_boxes[gidx * 4 + 2];
    const float b3 = gt_boxes[gidx * 4 + 3];
    const int x1 = min(max((int)b0, 0), IMG_W - 1);
    const int y1 = min(max((int)b1, 0), IMG_H - 1);
    const int x2 = max(x1 + 1, min((int)b2, IMG_W));
    const int y2 = max(y1 + 1, min((int)b3, IMG_H));
    const int cwi = x2 - x1, chi = y2 - y1;
    const float cw = (float)cwi, ch = (float)chi;

    const float* __restrict__ m   = gt_masks + (size_t)gidx * IMG_H * IMG_W;
    const float* __restrict__ lg1 = mask_logits + ((size_t)p * 2 + 1) * SS; // channel 1
    const float* __restrict__ lgA = mask_logits + (size_t)p * 2 * SS;       // both channels

    const v2f ones = {1.0f, 1.0f};

    // ---- BCE*pos reduction: 784 values = 12 full iters x 64 + masked tail --
    v8f accb0 = {}, accb1 = {};
    for (int it = 0; it < 12; it += 2) {
        v2f a0, a1;
        {
            const int base = it * 64 + lane * 2;
            a0[0] = bce_pixel(base,     m, lg1, x1, y1, cwi, chi, cw, ch, posw);
            a0[1] = bce_pixel(base + 1, m, lg1, x1, y1, cwi, chi, cw, ch, posw);
        }
        {
            const int base = (it + 1) * 64 + lane * 2;
            a1[0] = bce_pixel(base,     m, lg1, x1, y1, cwi, chi, cw, ch, posw);
            a1[1] = bce_pixel(base + 1, m, lg1, x1, y1, cwi, chi, cw, ch, posw);
        }
        accb0 = __builtin_amdgcn_wmma_f32_16x16x4_f32(false, a0, false, ones, (short)0, accb0, false, false);
        accb1 = __builtin_amdgcn_wmma_f32_16x16x4_f32(false, a1, false, ones, (short)0, accb1, false, false);
    }
    {   // tail: idx in [768, 832); valid while idx < 784. Branch-free:
        // clamp the index for addressing, zero the value by multiply-mask.
        v2f a0;
        const int base = 12 * 64 + lane * 2;
#pragma unroll
        for (int k = 0; k < 2; ++k) {
            const int   idx  = base + k;
            const int   idxc = min(idx, SS - 1);
            const float msk  = (idx < SS) ? 1.0f : 0.0f;
            a0[k] = bce_pixel(idxc, m, lg1, x1, y1, cwi, chi, cw, ch, posw) * msk;
        }
        accb0 = __builtin_amdgcn_wmma_f32_16x16x4_f32(false, a0, false, ones, (short)0, accb0, false, false);
    }

    // ---- sum(logits^2): 1568 values, float4 loads, 2 wmmas / iteration ----
    // 12 full iterations x 128 values + 1 masked tail (32 values).
    v8f accq0 = {}, accq1 = {};
    for (int it = 0; it < 12; ++it) {
        const v4f q = *(const v4f*)(lgA + it * 128 + lane * 4);
        v2f a0, a1;
        a0[0] = q.x * q.x; a0[1] = q.y * q.y;
        a1[0] = q.z * q.z; a1[1] = q.w * q.w;
        accq0 = __builtin_amdgcn_wmma_f32_16x16x4_f32(false, a0, false, ones, (short)0, accq0, false, false);
        accq1 = __builtin_amdgcn_wmma_f32_16x16x4_f32(false, a1, false, ones, (short)0, accq1, false, false);
    }
    {   // tail: base in [1536, 1664); valid while base < 1568 (lanes 0..7).
        const int   base  = 12 * 128 + lane * 4;
        const int   basec = min(base, 2 * SS - 4);        // 1564, 16B aligned
        const float msk   = (base < 2 * SS) ? 1.0f : 0.0f;
        const v4f q = *(const v4f*)(lgA + basec);
        v2f a0, a1;
        a0[0] = q.x * q.x * msk; a0[1] = q.y * q.y * msk;
        a1[0] = q.z * q.z * msk; a1[1] = q.w * q.w * msk;
        accq0 = __builtin_amdgcn_wmma_f32_16x16x4_f32(false, a0, false, ones, (short)0, accq0, false, false);
        accq1 = __builtin_amdgcn_wmma_f32_16x16x4_f32(false, a1, false, ones, (short)0, accq1, false, false);
    }

    // Merge accumulator pairs, then reduce.
    // D[m][n] is the accumulated row-sum of row m, identical for every n.
    // Column n=0 lives in lanes 0 (m=0..7) and 16 (m=8..15), VGPRs 0..7.
    const v8f accb = accb0 + accb1;
    const v8f accq = accq0 + accq1;
    float lb = accb[0] + accb[1] + accb[2] + accb[3] + accb[4] + accb[5] + accb[6] + accb[7];
    float lq = accq[0] + accq[1] + accq[2] + accq[3] + accq[4] + accq[5] + accq[6] + accq[7];
    const float sb = __shfl(lb, 0, 32) + __shfl(lb, 16, 32);
    const float sq = __shfl(lq, 0, 32) + __shfl(lq, 16, 32);
    if (lane == 0) {
        bsum[p] = sb;
        qsum[p] = sq;
    }
}

// ---------------------------------------------------------------------------
// Kernel 3: deterministic tree reduction of the 512 per-proposal partials,
// final scalar select.
// ---------------------------------------------------------------------------
__global__ void __launch_bounds__(256)
finalize_kernel(const float* __restrict__ bsum,
                const float* __restrict__ qsum,
                const float* __restrict__ pos,
                float* __restrict__ out)
{
    __shared__ float sb[256], sq[256], sp[256];
    const int t = threadIdx.x;
    sb[t] = bsum[t] + bsum[t + 256];
    sq[t] = qsum[t] + qsum[t + 256];
    sp[t] = pos[t]  + pos[t + 256];
    __syncthreads();
    for (int s = 128; s > 0; s >>= 1) {
        if (t < s) { sb[t] += sb[t + s]; sq[t] += sq[t + s]; sp[t] += sp[t + s]; }
        __syncthreads();
    }
    if (t == 0) {
        const float npos     = sp[0];
        const float loss_pos = sb[0] / (fmaxf(npos, 1.0f) * (float)SS);
        const float loss_neg = sq[0] * (0.01f / (float)(NUM_PROP * 2 * SS));
        out[0] = (npos > 0.0f) ? loss_pos : loss_neg;
    }
}

// ---------------------------------------------------------------------------
extern "C" void kernel_launch(void* const* d_in, const int* in_sizes, int n_in,
                              void* d_out, int out_size, void* d_ws, size_t ws_size,
                              hipStream_t stream)
{
    const float* proposals   = (const float*)d_in[0];  // [512,4]
    const float* gt_boxes    = (const float*)d_in[1];  // [128,4]
    const float* gt_masks    = (const float*)d_in[2];  // [128,520,704]
    const float* mask_logits = (const float*)d_in[3];  // [512,2,28,28]
    float* out = (float*)d_out;                        // scalar

    // Workspace layout (8 KB total):
    float* bsum    = (float*)d_ws;                                            // 512 f
    float* qsum    = (float*)((char*)d_ws + NUM_PROP * sizeof(float));        // 512 f
    int*   matched = (int*)  ((char*)d_ws + 2 * NUM_PROP * sizeof(float));    // 512 i
    float* pos     = (float*)((char*)d_ws + 3 * NUM_PROP * sizeof(float));    // 512 f

    iou_match_kernel<<<2, 256, 0, stream>>>(proposals, gt_boxes, matched, pos);
    mask_loss_kernel<<<NUM_PROP, 32, 0, stream>>>(gt_boxes, gt_masks, mask_logits,
                                                  matched, pos, bsum, qsum);
    finalize_kernel<<<1, 256, 0, stream>>>(bsum, qsum, pos, out);
}